// MyNet_37933151158488
// MI455X (gfx1250) — compile-verified
//
#include <hip/hip_runtime.h>
#include <math.h>

typedef __attribute__((ext_vector_type(2))) float v2f;
typedef __attribute__((ext_vector_type(8))) float v8f;

#define HID   25
#define NLET  31
#define KD    56            // NLET + HID (shared K for both layers)
#define KSTEPS 14           // 56 / 4
#define KPAIRS 28           // 56 / 2
#define NP1   112           // 4*HID=100 padded to 7*16
#define NT1   7
#define NP2   128           // 4*NLET=124 padded to 8*16
#define NT2   8
#define ROWS_PER_WAVE  16
#define WAVES_PER_BLOCK 8
#define ROWS_PER_BLOCK 128
#define CSTR  58            // comb LDS row stride (pad vs 56 for bank spread, even for b64)

// ---- LDS layout (float offsets) ----
#define OFF_WT1   0
#define SZ_WT1    (KPAIRS*NP1*2)                  // 6272
#define OFF_WT2   (OFF_WT1 + SZ_WT1)
#define SZ_WT2    (KPAIRS*NP2*2)                  // 7168
#define OFF_B1    (OFF_WT2 + SZ_WT2)              // 112
#define OFF_B2    (OFF_B1 + NP1)                  // 128
#define OFF_C1    (OFF_B2 + NP2)                  // 32 (pad)
#define OFF_C2    (OFF_C1 + 32)                   // 32 (pad)
#define OFF_COMB  (OFF_C2 + 32)
#define SZ_COMB   (ROWS_PER_BLOCK*CSTR)           // 7424
#define OFF_GATES (OFF_COMB + SZ_COMB)
#define SZ_GATES  (WAVES_PER_BLOCK*ROWS_PER_WAVE*NP2)  // 16384
#define SMEM_FLOATS (OFF_GATES + SZ_GATES)        // 37552 floats = 150208 B

__device__ __forceinline__ float sigmoidf_(float x) {
    return 1.0f / (1.0f + expf(-x));
}

__global__ void lstm2_wmma_kernel(const float* __restrict__ input,
                                  const float* __restrict__ output_in,
                                  const float* __restrict__ hidden_in,
                                  const float* __restrict__ context_in,
                                  const float* __restrict__ w1,
                                  const float* __restrict__ b1,
                                  const float* __restrict__ w2,
                                  const float* __restrict__ b2,
                                  float* __restrict__ out0,   // log_softmax (B,31)
                                  float* __restrict__ out1,   // hid (B,25)
                                  float* __restrict__ out2,   // ctx (B,56)
                                  int Bv) {
    extern __shared__ float smem[];
    float* sWt1  = smem + OFF_WT1;
    float* sWt2  = smem + OFF_WT2;
    float* sB1   = smem + OFF_B1;
    float* sB2   = smem + OFF_B2;
    float* sC1   = smem + OFF_C1;
    float* sC2   = smem + OFF_C2;
    float* sComb = smem + OFF_COMB;
    float* sGate = smem + OFF_GATES;

    const int tid  = threadIdx.x;
    const int lane = tid & 31;
    const int wid  = tid >> 5;
    const int h    = lane >> 4;     // half-wave id (0/1)
    const int m    = lane & 15;     // row/col within tile
    const int blockRow0 = blockIdx.x * ROWS_PER_BLOCK;
    const int rowBase   = blockRow0 + wid * ROWS_PER_WAVE;

    // ---- Cooperative LDS staging ----
    // W1^T, K-pair interleaved: sWt1[((K/2)*NP1 + n)*2 + (K&1)] = (n<100) ? w1[n*56+K] : 0
    for (int idx = tid; idx < NP1 * KD; idx += blockDim.x) {
        int n = idx / KD, K = idx - n * KD;
        float v = (n < 4 * HID) ? w1[n * KD + K] : 0.0f;
        sWt1[(((K >> 1) * NP1) + n) * 2 + (K & 1)] = v;
    }
    for (int idx = tid; idx < NP2 * KD; idx += blockDim.x) {
        int n = idx / KD, K = idx - n * KD;
        float v = (n < 4 * NLET) ? w2[n * KD + K] : 0.0f;
        sWt2[(((K >> 1) * NP2) + n) * 2 + (K & 1)] = v;
    }
    for (int idx = tid; idx < NP1; idx += blockDim.x)
        sB1[idx] = (idx < 4 * HID) ? b1[idx] : 0.0f;
    for (int idx = tid; idx < NP2; idx += blockDim.x)
        sB2[idx] = (idx < 4 * NLET) ? b2[idx] : 0.0f;
    if (tid < HID)  sC1[tid] = context_in[tid];          // context_in[0, :25]
    if (tid < NLET) sC2[tid] = context_in[HID + tid];    // context_in[0, 25:]

    // comb1 = [input(31) | hidden(25)] for this block's 128 rows
    for (int idx = tid; idx < ROWS_PER_BLOCK * KD; idx += blockDim.x) {
        int row = idx / KD, col = idx - row * KD;
        int gRow = blockRow0 + row;
        float v = 0.0f;
        if (gRow < Bv)
            v = (col < NLET) ? input[(size_t)gRow * NLET + col]
                             : hidden_in[(size_t)gRow * HID + (col - NLET)];
        sComb[row * CSTR + col] = v;
    }
    __syncthreads();

    float* sCombW = sComb + (wid * ROWS_PER_WAVE) * CSTR;
    float* sGateW = sGate + wid * (ROWS_PER_WAVE * NP2);

    // ================= Layer 1 GEMM: gates1 = comb1 @ W1^T =================
    {
        v2f a[KSTEPS];
        #pragma unroll
        for (int kk = 0; kk < KSTEPS; ++kk)
            a[kk] = *(const v2f*)&sCombW[m * CSTR + kk * 4 + 2 * h];

        for (int nt = 0; nt < NT1; ++nt) {
            v8f acc = {};
            #pragma unroll
            for (int kk = 0; kk < KSTEPS; ++kk) {
                v2f bfr = *(const v2f*)&sWt1[((2 * kk + h) * NP1 + nt * 16 + m) * 2];
                acc = __builtin_amdgcn_wmma_f32_16x16x4_f32(
                    false, a[kk], false, bfr, (short)0, acc, false, false);
            }
            int n = nt * 16 + m;
            float bias = sB1[n];
            #pragma unroll
            for (int r = 0; r < 8; ++r)
                sGateW[(r + 8 * h) * NP2 + n] = acc[r] + bias;
        }
    }
    __syncthreads();

    // ---- Layer 1 activations: ctx1, hid; build comb2 = [hid | output_in] ----
    for (int t = lane; t < ROWS_PER_WAVE * HID; t += 32) {
        int row = t / HID, j = t - row * HID;
        const float* g = &sGateW[row * NP2];
        float fg = sigmoidf_(g[j]);
        float ig = sigmoidf_(g[HID + j]);
        float cg = tanhf(g[2 * HID + j]);
        float og = sigmoidf_(g[3 * HID + j]);
        float ctx = sC1[j] * fg + ig * cg;
        float hid = tanhf(ctx) * og;
        sCombW[row * CSTR + j] = hid;
        int gRow = rowBase + row;
        if (gRow < Bv) {
            out1[(size_t)gRow * HID + j] = hid;
            out2[(size_t)gRow * KD + j] = ctx;
        }
    }
    for (int t = lane; t < ROWS_PER_WAVE * NLET; t += 32) {
        int row = t / NLET, j = t - row * NLET;
        int gRow = rowBase + row;
        sCombW[row * CSTR + HID + j] =
            (gRow < Bv) ? output_in[(size_t)gRow * NLET + j] : 0.0f;
    }
    __syncthreads();

    // ================= Layer 2 GEMM: gates2 = comb2 @ W2^T =================
    {
        v2f a[KSTEPS];
        #pragma unroll
        for (int kk = 0; kk < KSTEPS; ++kk)
            a[kk] = *(const v2f*)&sCombW[m * CSTR + kk * 4 + 2 * h];

        for (int nt = 0; nt < NT2; ++nt) {
            v8f acc = {};
            #pragma unroll
            for (int kk = 0; kk < KSTEPS; ++kk) {
                v2f bfr = *(const v2f*)&sWt2[((2 * kk + h) * NP2 + nt * 16 + m) * 2];
                acc = __builtin_amdgcn_wmma_f32_16x16x4_f32(
                    false, a[kk], false, bfr, (short)0, acc, false, false);
            }
            int n = nt * 16 + m;
            float bias = sB2[n];
            #pragma unroll
            for (int r = 0; r < 8; ++r)
                sGateW[(r + 8 * h) * NP2 + n] = acc[r] + bias;
        }
    }
    __syncthreads();

    // ---- Layer 2 activations: ctx2, out (pre-softmax into comb slots) ----
    for (int t = lane; t < ROWS_PER_WAVE * NLET; t += 32) {
        int row = t / NLET, j = t - row * NLET;
        const float* g = &sGateW[row * NP2];
        float fg = sigmoidf_(g[j]);
        float ig = sigmoidf_(g[NLET + j]);
        float cg = tanhf(g[2 * NLET + j]);
        float og = sigmoidf_(g[3 * NLET + j]);
        float ctx = sC2[j] * fg + ig * cg;
        float ov = tanhf(ctx) * og;
        sCombW[row * CSTR + j] = ov;
        int gRow = rowBase + row;
        if (gRow < Bv)
            out2[(size_t)gRow * KD + HID + j] = ctx;
    }
    __syncthreads();

    // ---- log_softmax over 31 logits per row; lanes 0..15 handle one row each ----
    if (lane < 16) {
        int row = lane;
        int gRow = rowBase + row;
        const float* v = &sCombW[row * CSTR];
        float mx = -INFINITY;
        #pragma unroll
        for (int j = 0; j < NLET; ++j) mx = fmaxf(mx, v[j]);
        float s = 0.0f;
        #pragma unroll
        for (int j = 0; j < NLET; ++j) s += expf(v[j] - mx);
        float ls = logf(s);
        if (gRow < Bv) {
            #pragma unroll
            for (int j = 0; j < NLET; ++j)
                out0[(size_t)gRow * NLET + j] = v[j] - mx - ls;
        }
    }
}

extern "C" void kernel_launch(void* const* d_in, const int* in_sizes, int n_in,
                              void* d_out, int out_size, void* d_ws, size_t ws_size,
                              hipStream_t stream) {
    const float* input      = (const float*)d_in[0];
    const float* output_in  = (const float*)d_in[1];
    const float* hidden_in  = (const float*)d_in[2];
    const float* context_in = (const float*)d_in[3];
    const float* w1         = (const float*)d_in[4];
    const float* b1         = (const float*)d_in[5];
    const float* w2         = (const float*)d_in[6];
    const float* b2         = (const float*)d_in[7];

    int Bv = in_sizes[0] / NLET;

    float* out = (float*)d_out;
    float* out0 = out;                                   // (B, 31)
    float* out1 = out + (size_t)Bv * NLET;               // (B, 25)
    float* out2 = out + (size_t)Bv * (NLET + HID);       // (B, 56)

    int nblocks = (Bv + ROWS_PER_BLOCK - 1) / ROWS_PER_BLOCK;
    size_t smem_bytes = (size_t)SMEM_FLOATS * sizeof(float);

    lstm2_wmma_kernel<<<nblocks, 256, smem_bytes, stream>>>(
        input, output_in, hidden_in, context_in, w1, b1, w2, b2,
        out0, out1, out2, Bv);
}